// DRGATAN_67104569033154
// MI455X (gfx1250) — compile-verified
//
#include <hip/hip_runtime.h>
#include <hip/hip_bf16.h>
#include <math.h>

// ---------------- problem constants ----------------
#define NN 10000
#define EE 160000

typedef __attribute__((ext_vector_type(2))) float v2f;
typedef __attribute__((ext_vector_type(8))) float v8f;

__device__ __forceinline__ float lrelu_(float x) { return x > 0.f ? x : 0.2f * x; }
__device__ __forceinline__ float elu_(float x)   { return x > 0.f ? x : (expf(x) - 1.f); }
// monotonic float<->uint mapping for atomicMax on floats
__device__ __forceinline__ unsigned fenc_(float f) {
    unsigned u = __float_as_uint(f);
    return (u & 0x80000000u) ? ~u : (u | 0x80000000u);
}
__device__ __forceinline__ float fdec_(unsigned e) {
    return __uint_as_float((e & 0x80000000u) ? (e & 0x7FFFFFFFu) : ~e);
}

// ---------------- dense per-relation GEMM via f32 WMMA ----------------
// Y[r, m, n] = sum_k X[m,k] * W[r,k,n]     (M % 16 == 0, K % 4 == 0)
// block = 128 threads (4 wave32s); block computes 16(M) x 64(N) for r = blockIdx.z
__global__ __launch_bounds__(128)
void k_gemm_rel_wmma(const float* __restrict__ X, const float* __restrict__ W,
                     float* __restrict__ Y, int M, int K, int Nout) {
    __shared__ float As[16 * 256];                 // A tile 16 x K (K <= 256)
    const int r    = blockIdx.z;
    const int m0   = blockIdx.x * 16;
    const int wave = threadIdx.x >> 5;
    const int lane = threadIdx.x & 31;
    const int n0   = blockIdx.y * 64 + wave * 16;

    // cooperative stage of A tile into LDS (shared across the 4 waves)
    for (int row = 0; row < 16; ++row)
        for (int col = threadIdx.x; col < K; col += 128)
            As[row * K + col] = X[(size_t)(m0 + row) * K + col];
    __syncthreads();

    if (n0 < Nout) {                                // uniform per-wave guard
        const float* Wr  = W + (size_t)r * K * Nout;
        const int  ncol  = n0 + (lane & 15);
        const bool nv    = ncol < Nout;
        const int  ncl   = nv ? ncol : (Nout - 1);  // clamp: branchless B loads
        const int  kh    = (lane >> 4) * 2;         // lanes 16-31 hold K+2,K+3
        const int  mrow  = lane & 15;
        v8f c = {};
        for (int k0 = 0; k0 < K; k0 += 4) {
            v2f a, b;
            a.x = As[mrow * K + k0 + kh];
            a.y = As[mrow * K + k0 + kh + 1];
            b.x = Wr[(size_t)(k0 + kh)     * Nout + ncl];
            b.y = Wr[(size_t)(k0 + kh + 1) * Nout + ncl];
            c = __builtin_amdgcn_wmma_f32_16x16x4_f32(false, a, false, b,
                                                      (short)0, c, false, false);
        }
        if (nv) {
            float* Yr = Y + (size_t)r * M * Nout;
            const int rbase = m0 + (lane >> 4) * 8;
#pragma unroll
            for (int i = 0; i < 8; ++i)
                Yr[(size_t)(rbase + i) * Nout + ncol] = c[i];
        }
    }
}

// ---------------- per-(relation,node) row dots: q_all / k_all ----------------
// one wave32 per row of XA[total, D]; H in {1,2}
__global__ __launch_bounds__(128)
void k_rowdot(const float* __restrict__ XA, const float* __restrict__ Q,
              const float* __restrict__ Kv, float* __restrict__ qall,
              float* __restrict__ kall, int D, int H, int total) {
    const int wid  = (int)((blockIdx.x * (unsigned)blockDim.x + threadIdx.x) >> 5);
    const int lane = threadIdx.x & 31;
    if (wid >= total) return;
    const float* xrow = XA + (size_t)wid * D;
    float aq[2] = {0.f, 0.f}, ak[2] = {0.f, 0.f};
    for (int d = lane; d < D; d += 32) {
        float v = xrow[d];
        for (int h = 0; h < H; ++h) {
            aq[h] += v * Q[d * H + h];
            ak[h] += v * Kv[d * H + h];
        }
    }
    for (int off = 16; off; off >>= 1)
        for (int h = 0; h < H; ++h) {
            aq[h] += __shfl_xor(aq[h], off, 32);
            ak[h] += __shfl_xor(ak[h], off, 32);
        }
    if (lane == 0)
        for (int h = 0; h < H; ++h) {
            qall[(size_t)wid * H + h] = aq[h];
            kall[(size_t)wid * H + h] = ak[h];
        }
}

// ---------------- init max/den/aggr buffers ----------------
__global__ void k_init(unsigned* __restrict__ menc, float* __restrict__ den,
                       float* __restrict__ aggr, int nh, int na) {
    const int stride = gridDim.x * blockDim.x;
    for (int t = blockIdx.x * blockDim.x + threadIdx.x; t < nh; t += stride) {
        menc[t] = 0x007FFFFFu;   // fenc(-inf)
        den[t]  = 0.f;
    }
    for (int t = blockIdx.x * blockDim.x + threadIdx.x; t < na; t += stride)
        aggr[t] = 0.f;
}

// ---------------- edge phase 1: alpha + segment max ----------------
__global__ void k_edge_alpha_max(const float* __restrict__ qall,
                                 const float* __restrict__ kall,
                                 const int* __restrict__ src, const int* __restrict__ dst,
                                 const int* __restrict__ etype,
                                 float* __restrict__ exbuf, unsigned* __restrict__ menc,
                                 int H, int Rr) {
    const int t = blockIdx.x * blockDim.x + threadIdx.x;
    if (t >= EE * H) return;
    const int e = t / H, h = t - e * H;
    const int et = (Rr > 1) ? etype[e] : 0;
    const int s = src[e], d = dst[e];
    const float a = lrelu_(qall[((size_t)et * NN + d) * H + h] +
                           kall[((size_t)et * NN + s) * H + h]);
    exbuf[t] = a;
    atomicMax(&menc[d * H + h], fenc_(a));
}

// ---------------- edge phase 2: exp + segment sum ----------------
__global__ void k_edge_exp(const int* __restrict__ dst, float* __restrict__ exbuf,
                           const unsigned* __restrict__ menc, float* __restrict__ den,
                           int H) {
    const int t = blockIdx.x * blockDim.x + threadIdx.x;
    if (t >= EE * H) return;
    const int e = t / H, h = t - e * H;
    const int d = dst[e];
    const float ex = expf(exbuf[t] - fdec_(menc[d * H + h]));
    exbuf[t] = ex;
    atomicAdd(&den[d * H + h], ex);
}

// ---------------- edge phase 3: weighted scatter-add (head mean folded in) ---
// one block per edge; threads cover output channels
__global__ void k_edge_aggr(const float* __restrict__ XA, const float* __restrict__ exbuf,
                            const float* __restrict__ den,
                            const int* __restrict__ src, const int* __restrict__ dst,
                            const int* __restrict__ etype, float* __restrict__ aggr,
                            int D, int C, int H, int Rr) {
    const int e = blockIdx.x;
    const int c = threadIdx.x;
    if (c >= C) return;
    const int s = src[e], d = dst[e];
    const int et = (Rr > 1) ? etype[e] : 0;
    const float* xrow = XA + ((size_t)et * NN + s) * D;   // L2-resident gather
    float v = 0.f;
    for (int h = 0; h < H; ++h) {
        const float attn = exbuf[(size_t)e * H + h] / (den[d * H + h] + 1e-16f);
        v += attn * xrow[h * C + c];
    }
    atomicAdd(&aggr[(size_t)d * C + c], v / (float)H);
}

// ---------------- node epilogue: elu(aggr + bias) ----------------
__global__ void k_epilogue(const float* __restrict__ aggr, const float* __restrict__ bias,
                           float* __restrict__ out, int C, int total) {
    const int t = blockIdx.x * blockDim.x + threadIdx.x;
    if (t >= total) return;
    out[t] = elu_(aggr[t] + bias[t % C]);
}

// ---------------- host-side conv driver ----------------
struct WsPtrs {
    float* xall; float* qall; float* kall; unsigned* menc;
    float* den;  float* exbuf; float* aggr;
};

static void run_attn_conv(const float* xin, const float* w, const float* qv,
                          const float* kv, const float* bias,
                          int K, int D, int C, int H, int Rr,
                          const int* src, const int* dst, const int* etype,
                          const WsPtrs& p, float* out, hipStream_t stream) {
    dim3 gg(NN / 16, (D + 63) / 64, Rr);
    k_gemm_rel_wmma<<<gg, 128, 0, stream>>>(xin, w, p.xall, NN, K, D);

    const int totalRN = Rr * NN;
    k_rowdot<<<(totalRN + 3) / 4, 128, 0, stream>>>(p.xall, qv, kv, p.qall, p.kall,
                                                    D, H, totalRN);
    k_init<<<512, 256, 0, stream>>>(p.menc, p.den, p.aggr, NN * H, NN * C);

    const int eh = EE * H;
    k_edge_alpha_max<<<(eh + 255) / 256, 256, 0, stream>>>(p.qall, p.kall, src, dst,
                                                           etype, p.exbuf, p.menc, H, Rr);
    k_edge_exp<<<(eh + 255) / 256, 256, 0, stream>>>(dst, p.exbuf, p.menc, p.den, H);

    const int bC = ((C + 63) / 64) * 64;
    k_edge_aggr<<<EE, bC, 0, stream>>>(p.xall, p.exbuf, p.den, src, dst, etype,
                                       p.aggr, D, C, H, Rr);
    k_epilogue<<<(NN * C + 255) / 256, 256, 0, stream>>>(p.aggr, bias, out, C, NN * C);
}

extern "C" void kernel_launch(void* const* d_in, const int* in_sizes, int n_in,
                              void* d_out, int out_size, void* d_ws, size_t ws_size,
                              hipStream_t stream) {
    (void)in_sizes; (void)n_in; (void)out_size; (void)ws_size;
    const float* x  = (const float*)d_in[0];
    const int*   ei = (const int*)  d_in[1];
    const int*   et = (const int*)  d_in[2];
    const float* w1 = (const float*)d_in[3],  *q1 = (const float*)d_in[4];
    const float* k1 = (const float*)d_in[5],  *b1 = (const float*)d_in[6];
    const float* wS = (const float*)d_in[7],  *qS = (const float*)d_in[8];
    const float* kS = (const float*)d_in[9],  *bS = (const float*)d_in[10];
    const float* w2 = (const float*)d_in[11], *q2 = (const float*)d_in[12];
    const float* k2 = (const float*)d_in[13], *b2 = (const float*)d_in[14];
    const float* wT = (const float*)d_in[15], *qT = (const float*)d_in[16];
    const float* kT = (const float*)d_in[17], *bT = (const float*)d_in[18];
    const float* wg = (const float*)d_in[19];
    const float* ag_src = (const float*)d_in[20];
    const float* ag_dst = (const float*)d_in[21];
    const float* bg = (const float*)d_in[22];

    const int* src = ei;        // edge_index[0]
    const int* dst = ei + EE;   // edge_index[1]

    // workspace layout (floats)
    float* ws = (float*)d_ws;
    size_t off = 0;
    WsPtrs p;
    p.xall  = ws + off;            off += (size_t)4 * NN * 512;  // 20,480,000 (82 MB, L2-resident)
    p.qall  = ws + off;            off += (size_t)4 * NN * 2;    //     80,000
    p.kall  = ws + off;            off += (size_t)4 * NN * 2;    //     80,000
    p.menc  = (unsigned*)(ws + off); off += (size_t)NN * 2;      //     20,000
    p.den   = ws + off;            off += (size_t)NN * 2;        //     20,000
    p.exbuf = ws + off;            off += (size_t)EE * 2;        //    320,000
    p.aggr  = ws + off;            off += (size_t)NN * 256;      //  2,560,000
    float* xs = ws + off;          off += (size_t)NN * 256;
    float* xt = ws + off;          off += (size_t)NN * 256;

    float* x_in   = (float*)d_out;
    float* x_out  = (float*)d_out + (size_t)NN * 65;
    float* x_self = (float*)d_out + (size_t)2 * NN * 65;

    // source branch: conv1 -> x_s -> convS -> x_in
    run_attn_conv(x,  w1, q1, k1, b1, 128, 512, 256, 2, 4, src, dst, et, p, xs,   stream);
    run_attn_conv(xs, wS, qS, kS, bS, 256, 130,  65, 2, 4, src, dst, et, p, x_in, stream);
    // target branch: conv2 -> x_t -> convT -> x_out
    run_attn_conv(x,  w2, q2, k2, b2, 128, 512, 256, 2, 4, src, dst, et, p, xt,    stream);
    run_attn_conv(xt, wT, qT, kT, bT, 256, 130,  65, 2, 4, src, dst, et, p, x_out, stream);
    // GAT self branch (R=1, H=1): logits = hs[src] + hd[dst] -> pass q:=ag_dst, k:=ag_src
    run_attn_conv(x,  wg, ag_dst, ag_src, bg, 128, 256, 256, 1, 1, src, dst, et, p, x_self, stream);
}